// iDecoder_1984274891165
// MI455X (gfx1250) — compile-verified
//
#include <hip/hip_runtime.h>
#include <hip/hip_bf16.h>
#include <math.h>

// ---- model dims (match reference) ----
#define N_LAYERS  16
#define DMODEL    1024
#define DINNER    2048
#define DSTATE    16
#define DCONV     4
#define DTRANK    64
#define NVOCAB2   32002
#define SEQL      2048
#define LN_EPS    1e-5f

typedef float v2f __attribute__((ext_vector_type(2)));
typedef float v8f __attribute__((ext_vector_type(8)));

// ---------------------------------------------------------------------
// CDNA5 async global->LDS copy (ASYNCcnt-tracked), 16 bytes per lane.
// VDST = LDS byte address VGPR, VADDR = 64-bit global address, saddr = off.
// ---------------------------------------------------------------------
__device__ __forceinline__ void async_ld_b128(unsigned lds_off, const float* gaddr) {
  asm volatile("global_load_async_to_lds_b128 %0, %1, off"
               :: "v"(lds_off), "v"(gaddr)
               : "memory");
}
__device__ __forceinline__ void wait_async_le3() {
  asm volatile("s_wait_asynccnt 3" ::: "memory");
}
__device__ __forceinline__ void wait_async_0() {
  asm volatile("s_wait_asynccnt 0" ::: "memory");
}
__device__ __forceinline__ unsigned lds_addr_of(const float* p) {
  // flat LDS address: low 32 bits are the wave-relative LDS byte offset
  return (unsigned)(size_t)p;
}

// =====================================================================
// GEMM: Y[M,N] = X[M,K](row stride lda) @ W[N,K]^T (+ bias)
// via V_WMMA_F32_16X16X4_F32, LDS double-buffered with async-to-LDS.
// Block: 256 thr (8 waves). Tile: 128(M) x 64(N). K chunk: 16.
// LDS row stride 20 floats -> 16B-aligned b128 stores, conflict-free
// ds_load_b64 fragment reads.
// Fragment layouts (ISA 7.12.2):
//   A 16x4 : lane holds {A[m][kb], A[m][kb+1]}, m=lane%16, kb=2*(lane/16)
//   B 4x16 : lane holds {W[n][kb], W[n][kb+1]},  n=lane%16
//   C/D    : lane l, vgpr v -> m = v + 8*(l/16), n = l%16
// Requires: M % 128 == 0, K % 32 == 0 (true at all call sites).
// =====================================================================
#define MBLK 128
#define NBLK 64
#define KC   16
#define LDSW 20   // padded row stride (floats)

__global__ __launch_bounds__(256)
void gemm_f32_wmma(const float* __restrict__ X, const float* __restrict__ W,
                   const float* __restrict__ bias, float* __restrict__ Y,
                   int M, int N, int K, int lda) {
  __shared__ float sA[2][MBLK * LDSW];
  __shared__ float sB[2][NBLK * LDSW];

  const int tid   = threadIdx.x;
  const int lane  = tid & 31;
  const int wave  = tid >> 5;           // 0..7
  const int half  = lane >> 4;          // 0/1
  const int lan16 = lane & 15;
  const int kb    = half * 2;           // K sub-offset {0,2}
  const int mblk  = blockIdx.x * MBLK;
  const int n0    = blockIdx.y * NBLK;

  // ---- cooperative load mapping (per K-chunk of 16) ----
  // A chunk: 128 rows x 16 cols -> thread: rowA = tid/2, c0A = (tid&1)*8 (2 x b128)
  // B chunk:  64 rows x 16 cols -> thread: rowB = tid/4, c0B = (tid&3)*4 (1 x b128)
  const int rowA = tid >> 1;
  const int c0A  = (tid & 1) * 8;
  const int rowB = tid >> 2;
  const int c0B  = (tid & 3) * 4;
  const int nB   = n0 + rowB;
  const int nrowB = (nB < N) ? nB : (N - 1);   // clamp: loads stay in bounds

  const float* gA = X + (size_t)(mblk + rowA) * lda + c0A;
  const float* gB = W + (size_t)nrowB * K + c0B;

  unsigned ldsA[2], ldsB[2];
  ldsA[0] = lds_addr_of(&sA[0][rowA * LDSW + c0A]);
  ldsA[1] = lds_addr_of(&sA[1][rowA * LDSW + c0A]);
  ldsB[0] = lds_addr_of(&sB[0][rowB * LDSW + c0B]);
  ldsB[1] = lds_addr_of(&sB[1][rowB * LDSW + c0B]);

  v8f acc0 = {}, acc1 = {}, acc2 = {}, acc3 = {};

  const float* As0 = &sA[0][(wave * 16 + lan16) * LDSW + kb];
  const float* As1 = &sA[1][(wave * 16 + lan16) * LDSW + kb];
  const float* Bs0 = &sB[0][lan16 * LDSW + kb];
  const float* Bs1 = &sB[1][lan16 * LDSW + kb];

  auto issue = [&](int chunk, int buf) {
    const float* gAc = gA + (size_t)chunk * KC;
    const float* gBc = gB + (size_t)chunk * KC;
    async_ld_b128(ldsA[buf],      gAc);
    async_ld_b128(ldsA[buf] + 16, gAc + 4);
    async_ld_b128(ldsB[buf],      gBc);
  };
  auto compute = [&](const float* As, const float* Bs) {
#pragma unroll
    for (int kc = 0; kc < 4; ++kc) {
      v2f a  = *(const v2f*)(As + kc * 4);
      v2f b0 = *(const v2f*)(Bs + kc * 4);
      v2f b1 = *(const v2f*)(Bs + 16 * LDSW + kc * 4);
      v2f b2 = *(const v2f*)(Bs + 32 * LDSW + kc * 4);
      v2f b3 = *(const v2f*)(Bs + 48 * LDSW + kc * 4);
      acc0 = __builtin_amdgcn_wmma_f32_16x16x4_f32(false, a, false, b0, (short)0, acc0, false, false);
      acc1 = __builtin_amdgcn_wmma_f32_16x16x4_f32(false, a, false, b1, (short)0, acc1, false, false);
      acc2 = __builtin_amdgcn_wmma_f32_16x16x4_f32(false, a, false, b2, (short)0, acc2, false, false);
      acc3 = __builtin_amdgcn_wmma_f32_16x16x4_f32(false, a, false, b3, (short)0, acc3, false, false);
    }
  };

  const int nchunks = K / KC;   // even at every call site (K % 32 == 0)

  // prologue: stream chunk 0 into buffer 0
  issue(0, 0);

  // branch-free paired main loop: processes chunks c (buf0) and c+1 (buf1),
  // always prefetching the next chunk.
  int c = 0;
  for (; c + 2 < nchunks; c += 2) {
    issue(c + 1, 1);
    wait_async_le3();
    __syncthreads();
    compute(As0, Bs0);          // chunk c from buf0
    __syncthreads();
    issue(c + 2, 0);
    wait_async_le3();
    __syncthreads();
    compute(As1, Bs1);          // chunk c+1 from buf1
    __syncthreads();
  }
  // epilogue: exactly two chunks remain (nchunks even)
  issue(c + 1, 1);
  wait_async_le3();
  __syncthreads();
  compute(As0, Bs0);
  __syncthreads();
  wait_async_0();
  __syncthreads();
  compute(As1, Bs1);

  // ---- store 16x64 strip per wave ----
  const int m0 = mblk + wave * 16;
  v8f accs[4] = {acc0, acc1, acc2, acc3};
#pragma unroll
  for (int t = 0; t < 4; ++t) {
    int n = n0 + t * 16 + lan16;
    if (n < N) {
      float bv = bias ? bias[n] : 0.0f;
#pragma unroll
      for (int v = 0; v < 8; ++v) {
        int m = m0 + v + 8 * half;
        Y[(size_t)m * N + n] = accs[t][v] + bv;
      }
    }
  }
}

// =====================================================================
// Fused residual-add + LayerNorm.  One block (256 thr) per row of 1024.
// resid = hid (+ resid if add_prev);  hs = LN(resid)*w + b
// =====================================================================
__global__ __launch_bounds__(256)
void ln_add_kernel(float* __restrict__ resid, const float* __restrict__ hid,
                   float* __restrict__ hs, const float* __restrict__ w,
                   const float* __restrict__ b, int add_prev) {
  const int row = blockIdx.x;
  const size_t base = (size_t)row * DMODEL;
  __shared__ float s_sum[8], s_sq[8];

  float v[4];
  float lsum = 0.f, lsq = 0.f;
#pragma unroll
  for (int i = 0; i < 4; ++i) {
    int idx = threadIdx.x + i * 256;
    float x = hid[base + idx];
    if (add_prev) x += resid[base + idx];
    resid[base + idx] = x;
    v[i] = x;
    lsum += x;
    lsq  += x * x;
  }
#pragma unroll
  for (int off = 16; off > 0; off >>= 1) {
    lsum += __shfl_xor(lsum, off, 32);
    lsq  += __shfl_xor(lsq,  off, 32);
  }
  int wv = threadIdx.x >> 5;
  if ((threadIdx.x & 31) == 0) { s_sum[wv] = lsum; s_sq[wv] = lsq; }
  __syncthreads();
  float tsum = 0.f, tsq = 0.f;
#pragma unroll
  for (int i = 0; i < 8; ++i) { tsum += s_sum[i]; tsq += s_sq[i]; }
  float mean = tsum * (1.0f / DMODEL);
  float var  = tsq * (1.0f / DMODEL) - mean * mean;
  float rstd = rsqrtf(var + LN_EPS);
#pragma unroll
  for (int i = 0; i < 4; ++i) {
    int idx = threadIdx.x + i * 256;
    hs[base + idx] = (v[i] - mean) * rstd * w[idx] + b[idx];
  }
}

// =====================================================================
// Causal depthwise conv (width 4) + bias + SiLU.
// Reads xs = xz[:, 0:DINNER] (row stride 2*DINNER); writes xs_act [L, DINNER].
// =====================================================================
__global__ __launch_bounds__(256)
void conv_silu_kernel(const float* __restrict__ xz, const float* __restrict__ cw,
                      const float* __restrict__ cb, float* __restrict__ xs_act) {
  int idx = blockIdx.x * blockDim.x + threadIdx.x;
  if (idx >= SEQL * DINNER) return;
  int d = idx & (DINNER - 1);
  int l = idx >> 11;                  // DINNER == 2048
  float acc = cb[d];
#pragma unroll
  for (int j = 0; j < DCONV; ++j) {
    int t = l - (DCONV - 1) + j;
    if (t >= 0) acc += cw[d * DCONV + j] * xz[(size_t)t * (2 * DINNER) + d];
  }
  xs_act[idx] = acc / (1.0f + __expf(-acc));   // silu
}

// =====================================================================
// Selective scan, fused: dt = softplus(dt_pre + dt_b);
//   h[s] = h[s]*exp(dt*A[s]) + dt*B_t[s]*u;  y = sum h[s]*C_t[s]
//   out  = (y + u*D) * silu(z)
// One thread per channel d (state of 16 in registers); sequential over L.
// =====================================================================
__global__ __launch_bounds__(256)
void scan_kernel(const float* __restrict__ xs_act, const float* __restrict__ dt_pre,
                 const float* __restrict__ dt_b,   const float* __restrict__ x_dbl,
                 const float* __restrict__ A_log,  const float* __restrict__ Dp,
                 const float* __restrict__ xz,     float* __restrict__ y_buf) {
  int d = blockIdx.x * blockDim.x + threadIdx.x;
  if (d >= DINNER) return;
  float A[DSTATE], h[DSTATE];
#pragma unroll
  for (int s = 0; s < DSTATE; ++s) {
    A[s] = -__expf(A_log[(size_t)d * DSTATE + s]);
    h[s] = 0.f;
  }
  const float Dval = Dp[d];
  const float dtb  = dt_b[d];
  for (int t = 0; t < SEQL; ++t) {
    float u   = xs_act[(size_t)t * DINNER + d];
    float dtp = dt_pre[(size_t)t * DINNER + d] + dtb;
    float dt  = (dtp > 20.f) ? dtp : __logf(1.f + __expf(dtp));   // softplus
    const float* bc = x_dbl + (size_t)t * (DTRANK + 2 * DSTATE) + DTRANK;
    float y = 0.f;
#pragma unroll
    for (int s = 0; s < DSTATE; ++s) {
      float dA = __expf(dt * A[s]);
      h[s] = h[s] * dA + (dt * bc[s]) * u;
      y += h[s] * bc[DSTATE + s];
    }
    float z  = xz[(size_t)t * (2 * DINNER) + DINNER + d];
    float sz = z / (1.f + __expf(-z));
    y_buf[(size_t)t * DINNER + d] = (y + u * Dval) * sz;
  }
}

// =====================================================================
extern "C" void kernel_launch(void* const* d_in, const int* in_sizes, int n_in,
                              void* d_out, int out_size, void* d_ws, size_t ws_size,
                              hipStream_t stream) {
  const float* x_in      = (const float*)d_in[0];
  const float* in_proj_w = (const float*)d_in[1];   // [16, 4096, 1024]
  const float* conv_w    = (const float*)d_in[2];   // [16, 2048, 4]
  const float* conv_b    = (const float*)d_in[3];   // [16, 2048]
  const float* x_proj_w  = (const float*)d_in[4];   // [16, 96, 2048]
  const float* dt_proj_w = (const float*)d_in[5];   // [16, 2048, 64]
  const float* dt_proj_b = (const float*)d_in[6];   // [16, 2048]
  const float* A_log     = (const float*)d_in[7];   // [16, 2048, 16]
  const float* Dmat      = (const float*)d_in[8];   // [16, 2048]
  const float* out_proj_w= (const float*)d_in[9];   // [16, 1024, 2048]
  const float* norm_w    = (const float*)d_in[10];  // [16, 1024]
  const float* norm_b    = (const float*)d_in[11];  // [16, 1024]
  const float* normf_w   = (const float*)d_in[12];  // [1024]
  const float* normf_b   = (const float*)d_in[13];  // [1024]
  const float* head_w    = (const float*)d_in[14];  // [32002, 1024]
  const float* head_b    = (const float*)d_in[15];  // [32002]
  float* out = (float*)d_out;

  // workspace layout (floats): ~26.2M floats (~105 MB)
  float* resid  = (float*)d_ws;
  float* hs     = resid  + (size_t)SEQL * DMODEL;
  float* hidden = hs     + (size_t)SEQL * DMODEL;
  float* xz     = hidden + (size_t)SEQL * DMODEL;
  float* xs_act = xz     + (size_t)SEQL * 2 * DINNER;
  float* x_dbl  = xs_act + (size_t)SEQL * DINNER;
  float* dt_pre = x_dbl  + (size_t)SEQL * (DTRANK + 2 * DSTATE);
  float* y_buf  = dt_pre + (size_t)SEQL * DINNER;

  const dim3 gblk(256);
  const int  gx = SEQL / MBLK;   // 16 M-tiles of 128 rows

  for (int i = 0; i < N_LAYERS; ++i) {
    const float* hid_in = (i == 0) ? x_in : hidden;
    ln_add_kernel<<<SEQL, 256, 0, stream>>>(
        resid, hid_in, hs, norm_w + (size_t)i * DMODEL, norm_b + (size_t)i * DMODEL, i != 0);

    // xz = hs @ in_proj_w^T   [2048,1024] x [4096,1024]^T -> [2048,4096]
    gemm_f32_wmma<<<dim3(gx, (2 * DINNER) / NBLK), gblk, 0, stream>>>(
        hs, in_proj_w + (size_t)i * 2 * DINNER * DMODEL, nullptr, xz,
        SEQL, 2 * DINNER, DMODEL, DMODEL);

    conv_silu_kernel<<<(SEQL * DINNER + 255) / 256, 256, 0, stream>>>(
        xz, conv_w + (size_t)i * DINNER * DCONV, conv_b + (size_t)i * DINNER, xs_act);

    // x_dbl = xs_act @ x_proj_w^T  [2048,2048] x [96,2048]^T -> [2048,96]
    gemm_f32_wmma<<<dim3(gx, (96 + NBLK - 1) / NBLK), gblk, 0, stream>>>(
        xs_act, x_proj_w + (size_t)i * (DTRANK + 2 * DSTATE) * DINNER, nullptr, x_dbl,
        SEQL, DTRANK + 2 * DSTATE, DINNER, DINNER);

    // dt_pre = x_dbl[:, :64] @ dt_proj_w^T  (lda = 96) -> [2048,2048]
    gemm_f32_wmma<<<dim3(gx, DINNER / NBLK), gblk, 0, stream>>>(
        x_dbl, dt_proj_w + (size_t)i * DINNER * DTRANK, nullptr, dt_pre,
        SEQL, DINNER, DTRANK, DTRANK + 2 * DSTATE);

    scan_kernel<<<DINNER / 256, 256, 0, stream>>>(
        xs_act, dt_pre, dt_proj_b + (size_t)i * DINNER, x_dbl,
        A_log + (size_t)i * DINNER * DSTATE, Dmat + (size_t)i * DINNER, xz, y_buf);

    // hidden = y_buf @ out_proj_w^T  [2048,2048] x [1024,2048]^T -> [2048,1024]
    gemm_f32_wmma<<<dim3(gx, DMODEL / NBLK), gblk, 0, stream>>>(
        y_buf, out_proj_w + (size_t)i * DMODEL * DINNER, nullptr, hidden,
        SEQL, DMODEL, DINNER, DINNER);
  }

  // final residual + LN
  ln_add_kernel<<<SEQL, 256, 0, stream>>>(resid, hidden, hs, normf_w, normf_b, 1);

  // logits = hs @ head_w^T + head_b  -> [2048, 32002]  (num_last_tokens == 0)
  gemm_f32_wmma<<<dim3(gx, (NVOCAB2 + NBLK - 1) / NBLK), gblk, 0, stream>>>(
      hs, head_w, head_b, out, SEQL, NVOCAB2, DMODEL, DMODEL);
}